// UniversalSpectralCF_46102178955759
// MI455X (gfx1250) — compile-verified
//
#include <hip/hip_runtime.h>
#include <math.h>

#define EPSF 1e-8f
#define NEG_INF -3.0e38f

typedef float v2f __attribute__((ext_vector_type(2)));
typedef float v8f __attribute__((ext_vector_type(8)));

static inline int cdiv_h(int a, int b) { return (a + b - 1) / b; }

// ---------------------------------------------------------------------------
// Row L2-normalize:  Rn[r][c] = R[r][c] / (||R[r]|| + EPS)
// ---------------------------------------------------------------------------
__global__ __launch_bounds__(256) void k_row_normalize(const float* __restrict__ R,
                                                       float* __restrict__ Rn,
                                                       int nrows, int ncols) {
  __shared__ float red[256];
  const int row = blockIdx.x;
  const int tid = threadIdx.x;
  const float* src = R + (size_t)row * ncols;
  float s = 0.f;
  for (int c = tid; c < ncols; c += 256) { float v = src[c]; s += v * v; }
  red[tid] = s;
  __syncthreads();
  for (int st = 128; st > 0; st >>= 1) {
    if (tid < st) red[tid] += red[tid + st];
    __syncthreads();
  }
  const float scale = 1.f / (sqrtf(red[0]) + EPSF);
  float* dst = Rn + (size_t)row * ncols;
  for (int c = tid; c < ncols; c += 256) dst[c] = src[c] * scale;
}

// ---------------------------------------------------------------------------
// Column inverse norms of R (nrows x ncols): cinv[c] = 1/(||R[:,c]|| + EPS)
// ---------------------------------------------------------------------------
__global__ __launch_bounds__(256) void k_col_inv_norms(const float* __restrict__ R,
                                                       float* __restrict__ cinv,
                                                       int nrows, int ncols) {
  const int c = blockIdx.x * 256 + threadIdx.x;
  if (c >= ncols) return;
  float s = 0.f;
  for (int u = 0; u < nrows; ++u) { float v = R[(size_t)u * ncols + c]; s += v * v; }
  cinv[c] = 1.f / (sqrtf(s) + EPSF);
}

// ---------------------------------------------------------------------------
// Rtn[i][u] = R[u][i] * cinv[i]   (tiled transpose, 32x32 via LDS)
// ---------------------------------------------------------------------------
__global__ __launch_bounds__(256) void k_transpose_scale(const float* __restrict__ R,
                                                         const float* __restrict__ cinv,
                                                         float* __restrict__ Rtn,
                                                         int nrows, int ncols) {
  __shared__ float tile[32][33];
  const int i0 = blockIdx.x * 32;
  const int u0 = blockIdx.y * 32;
  const int x = threadIdx.x;   // 0..31
  const int y = threadIdx.y;   // 0..7
#pragma unroll
  for (int r = 0; r < 4; ++r) {
    int u = u0 + y + r * 8;
    int i = i0 + x;
    tile[y + r * 8][x] = (u < nrows && i < ncols) ? R[(size_t)u * ncols + i] : 0.f;
  }
  __syncthreads();
#pragma unroll
  for (int r = 0; r < 4; ++r) {
    int i = i0 + y + r * 8;
    int u = u0 + x;
    if (i < ncols && u < nrows) Rtn[(size_t)i * nrows + u] = tile[x][y + r * 8] * cinv[i];
  }
}

// ---------------------------------------------------------------------------
// f32 WMMA GEMM:  C[M,N] = A[M,K] * op(B)      op(B)=B (K,N) or B^T (N,K)
// Block 256x128, BK=32, 256 threads = 8 waves (4Mx2N); wave tile 64x64 =
// 4x4 tiles of 16x16 -> 16 V_WMMA_F32_16X16X4_F32 per k-step, 128 acc VGPRs.
// B staged in K-group-of-4 interleave: Blds[((k>>2)*BN + n)*4 + (k&3)] so a
// B fragment is one aligned ds_load_b64 and lane-halves hit disjoint banks.
// ---------------------------------------------------------------------------
template <bool TRANSB>
__global__ __launch_bounds__(256) void k_gemm_f32_wmma(const float* __restrict__ A,
                                                       const float* __restrict__ B,
                                                       float* __restrict__ C,
                                                       int M, int N, int K,
                                                       int lda, int ldb, int ldc) {
  constexpr int BM = 256, BN = 128, BK = 32;
  constexpr int AST = BK + 2;   // 34 floats: conflict-free A-frag column reads
  __shared__ float Alds[BM * AST];
  __shared__ float Blds[BK * BN];   // K-group-of-4 interleaved, conflict-free

  const int tid  = threadIdx.x;
  const int lane = tid & 31;     // wave32
  const int wave = tid >> 5;     // 0..7
  const int wrow = wave >> 1;    // 0..3  -> 64-row slab
  const int wcol = wave & 1;     // 0..1  -> 64-col slab
  const int hl   = lane >> 4;    // lane half (0/1)
  const int l16  = lane & 15;
  const int bm = blockIdx.y * BM;
  const int bn = blockIdx.x * BN;

  const v8f vzero = {0.f, 0.f, 0.f, 0.f, 0.f, 0.f, 0.f, 0.f};
  v8f acc[4][4];
#pragma unroll
  for (int i = 0; i < 4; ++i)
#pragma unroll
    for (int j = 0; j < 4; ++j) acc[i][j] = vzero;

  for (int k0 = 0; k0 < K; k0 += BK) {
    // ---- stage A block (256 x 32): 2048 float4 slots / 256 threads ----
#pragma unroll
    for (int r = 0; r < 8; ++r) {
      int idx  = tid + r * 256;
      int arow = idx >> 3;
      int ak   = (idx & 7) << 2;
      int gr = bm + arow, gk = k0 + ak;
      float x0 = 0.f, x1 = 0.f, x2 = 0.f, x3 = 0.f;
      if (gr < M) {
        const float* p = A + (size_t)gr * lda + gk;
        if (gk + 3 < K) {
          float4 v = *(const float4*)p;
          x0 = v.x; x1 = v.y; x2 = v.z; x3 = v.w;
          if (k0 + BK < K) __builtin_prefetch(p + BK, 0, 1);  // global_prefetch_b8
        } else {
          if (gk + 0 < K) x0 = p[0];
          if (gk + 1 < K) x1 = p[1];
          if (gk + 2 < K) x2 = p[2];
          if (gk + 3 < K) x3 = p[3];
        }
      }
      float* q = &Alds[arow * AST + ak];
      q[0] = x0; q[1] = x1; q[2] = x2; q[3] = x3;
    }
    // ---- stage B block (32 x 128) into K-group-of-4 interleave ----
    if (!TRANSB) {
#pragma unroll
      for (int r = 0; r < 4; ++r) {
        int idx  = tid + r * 256;
        int brow = idx >> 5;            // k within block
        int bc   = (idx & 31) << 2;     // n within block
        int gk = k0 + brow, gn = bn + bc;
        float x0 = 0.f, x1 = 0.f, x2 = 0.f, x3 = 0.f;
        if (gk < K) {
          const float* p = B + (size_t)gk * ldb + gn;
          if (gn + 3 < N) {
            float4 v = *(const float4*)p;
            x0 = v.x; x1 = v.y; x2 = v.z; x3 = v.w;
          } else {
            if (gn + 0 < N) x0 = p[0];
            if (gn + 1 < N) x1 = p[1];
            if (gn + 2 < N) x2 = p[2];
            if (gn + 3 < N) x3 = p[3];
          }
        }
        int base = (((brow >> 2) * BN) + bc) * 4 + (brow & 3);
        Blds[base + 0]  = x0;
        Blds[base + 4]  = x1;
        Blds[base + 8]  = x2;
        Blds[base + 12] = x3;
      }
    } else {
      // B element (k,n) = Bsrc[n*ldb + k]; 4 consecutive k -> one b128 store
#pragma unroll
      for (int r = 0; r < 4; ++r) {
        int idx = tid + r * 256;
        int bnr = idx >> 3;             // n within block (0..127)
        int bk  = (idx & 7) << 2;       // k within block (multiple of 4)
        int gn = bn + bnr, gk = k0 + bk;
        float x0 = 0.f, x1 = 0.f, x2 = 0.f, x3 = 0.f;
        if (gn < N) {
          const float* p = B + (size_t)gn * ldb + gk;
          if (gk + 3 < K) {
            float4 v = *(const float4*)p;
            x0 = v.x; x1 = v.y; x2 = v.z; x3 = v.w;
          } else {
            if (gk + 0 < K) x0 = p[0];
            if (gk + 1 < K) x1 = p[1];
            if (gk + 2 < K) x2 = p[2];
            if (gk + 3 < K) x3 = p[3];
          }
        }
        float4 st; st.x = x0; st.y = x1; st.z = x2; st.w = x3;
        *(float4*)&Blds[(((bk >> 2) * BN) + bnr) * 4] = st;
      }
    }
    __syncthreads();

    // ---- compute: 8 k-steps x 16 WMMAs ----
#pragma unroll
    for (int kk = 0; kk < BK; kk += 4) {
      v2f a[4], b[4];
#pragma unroll
      for (int i = 0; i < 4; ++i) {
        // A 16x4 frag: lanes 0-15 hold (K0,K1), lanes 16-31 hold (K2,K3)
        int m = wrow * 64 + i * 16 + l16;
        const float* p = &Alds[m * AST + kk + 2 * hl];
        a[i][0] = p[0];
        a[i][1] = p[1];
      }
#pragma unroll
      for (int j = 0; j < 4; ++j) {
        // B 4x16 frag: one contiguous float2 per lane (single ds_load_b64)
        int nn = wcol * 64 + j * 16 + l16;
        const float* p = &Blds[(((kk >> 2) * BN) + nn) * 4 + 2 * hl];
        b[j][0] = p[0];
        b[j][1] = p[1];
      }
#pragma unroll
      for (int i = 0; i < 4; ++i)
#pragma unroll
        for (int j = 0; j < 4; ++j)
          acc[i][j] = __builtin_amdgcn_wmma_f32_16x16x4_f32(
              false, a[i], false, b[j], (short)0, acc[i][j], false, false);
    }
    __syncthreads();
  }

  // ---- store C: D layout: M = vgpr + 8*half, N = lane%16 ----
#pragma unroll
  for (int i = 0; i < 4; ++i) {
    int mt = bm + wrow * 64 + i * 16 + hl * 8;
#pragma unroll
    for (int j = 0; j < 4; ++j) {
      int nn = bn + wcol * 64 + j * 16 + l16;
      if (nn < N) {
#pragma unroll
        for (int r = 0; r < 8; ++r) {
          int m = mt + r;
          if (m < M) C[(size_t)m * ldc + nn] = acc[i][j][r];
        }
      }
    }
  }
}

// ---------------------------------------------------------------------------
// Per-row top-k adjacency (faithful to reference semantics):
// take top-(k+1) picks (ties -> lower index), drop self, keep first k,
// scatter into Adj row (rest zeros).
// ---------------------------------------------------------------------------
__global__ __launch_bounds__(256) void k_topk_row(const float* __restrict__ S,
                                                  float* __restrict__ Adj,
                                                  int n, const int* __restrict__ kptr) {
  __shared__ float srow[6016];
  __shared__ float rv[256];
  __shared__ int   ri[256];
  __shared__ float pv[64];
  __shared__ int   pi[64];
  __shared__ float kv[64];
  __shared__ int   ki[64];
  __shared__ int   kept_cnt;

  const int row = blockIdx.x;
  const int tid = threadIdx.x;
  int k = *kptr;
  if (k > 62) k = 62;
  if (k > n - 1) k = n - 1;
  int picks = k + 1;
  if (picks > n) picks = n;

  const float* src = S + (size_t)row * n;
  for (int c = tid; c < n; c += 256) srow[c] = src[c];
  __syncthreads();

  for (int p = 0; p < picks; ++p) {
    float bv = NEG_INF;
    int bi = 0x7fffffff;
    for (int c = tid; c < n; c += 256) {
      float v = srow[c];
      if (v > bv) { bv = v; bi = c; }
    }
    rv[tid] = bv; ri[tid] = bi;
    __syncthreads();
    for (int st = 128; st > 0; st >>= 1) {
      if (tid < st) {
        float ov = rv[tid + st]; int oi = ri[tid + st];
        if (ov > rv[tid] || (ov == rv[tid] && oi < ri[tid])) { rv[tid] = ov; ri[tid] = oi; }
      }
      __syncthreads();
    }
    if (tid == 0) { pv[p] = rv[0]; pi[p] = ri[0]; srow[ri[0]] = NEG_INF; }
    __syncthreads();
  }

  if (tid == 0) {
    int cnt = 0;
    for (int p = 0; p < picks && cnt < k; ++p)
      if (pi[p] != row) { kv[cnt] = pv[p]; ki[cnt] = pi[p]; ++cnt; }
    kept_cnt = cnt;
  }
  __syncthreads();

  const int cnt = kept_cnt;
  float* dst = Adj + (size_t)row * n;
  for (int c = tid; c < n; c += 256) {
    float v = 0.f;
    for (int q = 0; q < cnt; ++q)
      if (ki[q] == c) v = kv[q];
    dst[c] = v;
  }
}

// ---------------------------------------------------------------------------
// In-place symmetrize: A = (A + A^T) / 2   (each unordered pair once)
// ---------------------------------------------------------------------------
__global__ __launch_bounds__(256) void k_symmetrize(float* __restrict__ A, int n) {
  const int j = blockIdx.x * 16 + threadIdx.x;
  const int i = blockIdx.y * 16 + threadIdx.y;
  if (i < n && j < n && j >= i) {
    float a = A[(size_t)i * n + j];
    float b = A[(size_t)j * n + i];
    float m = 0.5f * (a + b);
    A[(size_t)i * n + j] = m;
    A[(size_t)j * n + i] = m;
  }
}

// ---------------------------------------------------------------------------
// Row sums -> inv[i] = rsqrt(sum_i + EPS)   (A symmetric: rowsum == colsum)
// ---------------------------------------------------------------------------
__global__ __launch_bounds__(256) void k_row_rsqrt(const float* __restrict__ A,
                                                   float* __restrict__ inv, int n) {
  __shared__ float red[256];
  const int row = blockIdx.x;
  const int tid = threadIdx.x;
  const float* src = A + (size_t)row * n;
  float s = 0.f;
  for (int c = tid; c < n; c += 256) s += src[c];
  red[tid] = s;
  __syncthreads();
  for (int st = 128; st > 0; st >>= 1) {
    if (tid < st) red[tid] += red[tid + st];
    __syncthreads();
  }
  if (tid == 0) inv[row] = rsqrtf(red[0] + EPSF);
}

// ---------------------------------------------------------------------------
// A[i][j] *= inv[i] * inv[j]
// ---------------------------------------------------------------------------
__global__ __launch_bounds__(256) void k_scale_sym(float* __restrict__ A,
                                                   const float* __restrict__ inv, int n) {
  const int j = blockIdx.x * 16 + threadIdx.x;
  const int i = blockIdx.y * 16 + threadIdx.y;
  if (i < n && j < n) A[(size_t)i * n + j] *= inv[i] * inv[j];
}

// ---------------------------------------------------------------------------
// Launcher
// ---------------------------------------------------------------------------
extern "C" void kernel_launch(void* const* d_in, const int* in_sizes, int n_in,
                              void* d_out, int out_size, void* d_ws, size_t ws_size,
                              hipStream_t stream) {
  const int NU = 6000, NI = 4000;
  const float* R  = (const float*)d_in[0];
  const int* kU   = (const int*)d_in[1];
  const int* kI   = (const int*)d_in[2];
  float* out = (float*)d_out;

  // workspace layout (floats)
  float* ws   = (float*)d_ws;
  float* bufN = ws;                          // 24,000,000  : Rn, later Rtn
  float* bufS = bufN + (size_t)NU * NI;      // 36,000,000  : S_u / S_i / T
  float* adjU = bufS + (size_t)NU * NU;      // 36,000,000
  float* adjI = adjU + (size_t)NU * NU;      // 16,000,000
  float* vecU = adjI + (size_t)NI * NI;      // 6000
  float* vecI = vecU + NU;                   // 4000
  (void)ws_size; (void)n_in; (void)in_sizes; (void)out_size;

  const dim3 blk256(256);
  const dim3 blkT(32, 8);
  const dim3 blk16(16, 16);

  // 1. user-normalized rows Rn
  k_row_normalize<<<NU, blk256, 0, stream>>>(R, bufN, NU, NI);

  // 2. S_u = Rn * Rn^T   (M=N=6000, K=4000)
  {
    dim3 grid(cdiv_h(NU, 128), cdiv_h(NU, 256));
    k_gemm_f32_wmma<true><<<grid, blk256, 0, stream>>>(bufN, bufN, bufS,
                                                       NU, NU, NI, NI, NI, NU);
  }

  // 3. user adjacency: top-k, symmetrize, sym-normalize
  k_topk_row<<<NU, blk256, 0, stream>>>(bufS, adjU, NU, kU);
  {
    dim3 grid(cdiv_h(NU, 16), cdiv_h(NU, 16));
    k_symmetrize<<<grid, blk16, 0, stream>>>(adjU, NU);
    k_row_rsqrt<<<NU, blk256, 0, stream>>>(adjU, vecU, NU);
    k_scale_sym<<<grid, blk16, 0, stream>>>(adjU, vecU, NU);
  }

  // 4. item-normalized columns -> Rtn (4000 x 6000)
  k_col_inv_norms<<<cdiv_h(NI, 256), blk256, 0, stream>>>(R, vecI, NU, NI);
  {
    dim3 grid(cdiv_h(NI, 32), cdiv_h(NU, 32));
    k_transpose_scale<<<grid, blkT, 0, stream>>>(R, vecI, bufN, NU, NI);
  }

  // 5. S_i = Rtn * Rtn^T   (M=N=4000, K=6000)
  {
    dim3 grid(cdiv_h(NI, 128), cdiv_h(NI, 256));
    k_gemm_f32_wmma<true><<<grid, blk256, 0, stream>>>(bufN, bufN, bufS,
                                                       NI, NI, NU, NU, NU, NI);
  }

  // 6. item adjacency
  k_topk_row<<<NI, blk256, 0, stream>>>(bufS, adjI, NI, kI);
  {
    dim3 grid(cdiv_h(NI, 16), cdiv_h(NI, 16));
    k_symmetrize<<<grid, blk16, 0, stream>>>(adjI, NI);
    k_row_rsqrt<<<NI, blk256, 0, stream>>>(adjI, vecI, NI);
    k_scale_sym<<<grid, blk16, 0, stream>>>(adjI, vecI, NI);
  }

  // 7. T = R * adjI   (M=6000, N=4000, K=4000) -> bufS
  {
    dim3 grid(cdiv_h(NI, 128), cdiv_h(NU, 256));
    k_gemm_f32_wmma<false><<<grid, blk256, 0, stream>>>(R, adjI, bufS,
                                                        NU, NI, NI, NI, NI, NI);
  }

  // 8. out = adjU * T   (M=6000, N=4000, K=6000)
  {
    dim3 grid(cdiv_h(NI, 128), cdiv_h(NU, 256));
    k_gemm_f32_wmma<false><<<grid, blk256, 0, stream>>>(adjU, bufS, out,
                                                        NU, NI, NU, NU, NI, NI);
  }
}